// TriangleRenderer_13597866459711
// MI455X (gfx1250) — compile-verified
//
#include <hip/hip_runtime.h>

typedef __attribute__((ext_vector_type(2))) float v2f;
typedef __attribute__((ext_vector_type(4))) float v4f;
typedef __attribute__((ext_vector_type(8))) float v8f;

#define NT   256
#define IH   512
#define IW   512
// -SHARPNESS * log2(e): fold sigmoid sharpness + exp->exp2 conversion into edge coeffs
#define NEG_S_LOG2E (-144.2695040888963f)
#define EPSV 1e-6f

__global__ __launch_bounds__(256) void tri_render_kernel(
    const float* __restrict__ verts,    // [256][3][2]
    const float* __restrict__ colors,   // [256][3]
    const float* __restrict__ depth,    // [256]
    float* __restrict__ out)            // [3][512][512]
{
    __shared__ float smem[NT * 16];

    const int tid = threadIdx.x;

    // ---- per-triangle constants -> LDS (one triangle per thread) ----
    {
        const int t = tid;
        const float ax = verts[t*6+0], ay = verts[t*6+1];
        const float bx = verts[t*6+2], by = verts[t*6+3];
        const float cx = verts[t*6+4], cy = verts[t*6+5];
        const float s = NEG_S_LOG2E;
        // edge(p;a,b) = (px-ax)(by-ay) - (py-ay)(bx-ax) = A*px + B*py + C
        float A0 = s*(by-ay), B0 = -s*(bx-ax), C0 = s*(ay*(bx-ax) - ax*(by-ay));
        float A1 = s*(cy-by), B1 = -s*(cx-bx), C1 = s*(by*(cx-bx) - bx*(cy-by));
        float A2 = s*(ay-cy), B2 = -s*(ax-cx), C2 = s*(cy*(ax-cx) - cx*(ay-cy));
        float ed = __expf(depth[t]);     // softmax shift-invariance: w = (mask+eps)*e^d
        float* p = &smem[t*16];
        p[0]=A0;  p[1]=B0;  p[2]=C0;  p[3]=A1;
        p[4]=B1;  p[5]=C1;  p[6]=A2;  p[7]=B2;
        p[8]=C2;  p[9]=ed;  p[10]=EPSV*ed; p[11]=colors[t*3+0];
        p[12]=colors[t*3+1]; p[13]=colors[t*3+2]; p[14]=0.f; p[15]=0.f;
    }
    __syncthreads();

    const int lane = tid & 31;
    const int wv   = tid >> 5;
    const int sel  = lane & 15;   // pixel row of A / column of B (WMMA layout)
    const int half = lane >> 4;   // 0: K=0,1  1: K=2,3
    const int y    = blockIdx.x;
    const float py = (float)y * (1.0f/511.0f);
    const int xbase = wv * 64;    // 8 waves * 64 px = one full image row

    float px[4];
    #pragma unroll
    for (int t = 0; t < 4; ++t)
        px[t] = (float)(xbase + 16*t + sel) * (1.0f/511.0f);

    // B column selectors: col0=r, col1=g, col2=b, col3=1 (weight-sum), rest 0
    const float mR = (sel==0)?1.f:0.f;
    const float mG = (sel==1)?1.f:0.f;
    const float mB = (sel==2)?1.f:0.f;
    const float mS = (sel==3)?1.f:0.f;

    v8f acc[4] = {};                       // 4 tiles of 16 px, D = 16x16 f32
    const v4f* lds4 = (const v4f*)smem;

    for (int k = 0; k < 64; ++k) {
        const int tri = 4*k + 2*half;      // this lane's first triangle in the group
        v4f q0 = lds4[tri*4+0];            // A0 B0 C0 A1
        v4f q1 = lds4[tri*4+1];            // B1 C1 A2 B2
        v4f q2 = lds4[tri*4+2];            // C2 expd eps*expd cr
        v4f q3 = lds4[tri*4+3];            // cg cb - -
        v4f r0 = lds4[tri*4+4];            // second triangle (tri+1)
        v4f r1 = lds4[tri*4+5];
        v4f r2 = lds4[tri*4+6];
        v4f r3 = lds4[tri*4+7];

        // hoist the uniform-y part of each edge
        const float qy0 = fmaf(q0.y, py, q0.z);
        const float qy1 = fmaf(q1.x, py, q1.y);
        const float qy2 = fmaf(q1.w, py, q2.x);
        const float ry0 = fmaf(r0.y, py, r0.z);
        const float ry1 = fmaf(r1.x, py, r1.y);
        const float ry2 = fmaf(r1.w, py, r2.x);

        v2f bcol;                          // B rows for this lane's 2 triangles
        bcol.x = fmaf(q2.w, mR, fmaf(q3.x, mG, fmaf(q3.y, mB, mS)));
        bcol.y = fmaf(r2.w, mR, fmaf(r3.x, mG, fmaf(r3.y, mB, mS)));

        #pragma unroll
        for (int t = 0; t < 4; ++t) {
            const float x = px[t];
            // w = rcp((1+e^-Se0)(1+e^-Se1)(1+e^-Se2)) * e^d + eps*e^d
            float u0 = __builtin_amdgcn_exp2f(fmaf(q0.x, x, qy0));
            float u1 = __builtin_amdgcn_exp2f(fmaf(q0.w, x, qy1));
            float u2 = __builtin_amdgcn_exp2f(fmaf(q1.z, x, qy2));
            float dq = (1.f+u0)*(1.f+u1)*(1.f+u2);
            float w0 = fmaf(__builtin_amdgcn_rcpf(dq), q2.y, q2.z);

            float s0 = __builtin_amdgcn_exp2f(fmaf(r0.x, x, ry0));
            float s1 = __builtin_amdgcn_exp2f(fmaf(r0.w, x, ry1));
            float s2 = __builtin_amdgcn_exp2f(fmaf(r1.z, x, ry2));
            float dr = (1.f+s0)*(1.f+s1)*(1.f+s2);
            float w1 = fmaf(__builtin_amdgcn_rcpf(dr), r2.y, r2.z);

            v2f amat; amat.x = w0; amat.y = w1;
            // D[pixel, {r,g,b,sum}] += A(16px x 4tri) * B(4tri x cols)
            acc[t] = __builtin_amdgcn_wmma_f32_16x16x4_f32(
                false, amat, false, bcol, (short)0, acc[t], false, false);
        }
    }

    // ---- epilogue: normalize by the sum column (N=3 -> lanes 3 / 19) ----
    const int sumLane = (lane & 16) | 3;
    #pragma unroll
    for (int t = 0; t < 4; ++t) {
        float res[8];
        #pragma unroll
        for (int r = 0; r < 8; ++r) {
            float sw = __shfl(acc[t][r], sumLane, 32);  // Σw for pixel row M
            res[r] = acc[t][r] / sw;
        }
        if (sel < 3) {
            float* o = out + sel*(IH*IW) + y*IW + xbase + t*16 + half*8;
            v4f lo = { res[0], res[1], res[2], res[3] };
            v4f hi = { res[4], res[5], res[6], res[7] };
            *(v4f*)(o)     = lo;
            *(v4f*)(o + 4) = hi;
        }
    }
}

extern "C" void kernel_launch(void* const* d_in, const int* in_sizes, int n_in,
                              void* d_out, int out_size, void* d_ws, size_t ws_size,
                              hipStream_t stream) {
    const float* verts  = (const float*)d_in[0];
    const float* colors = (const float*)d_in[1];
    const float* depth  = (const float*)d_in[2];
    float* out = (float*)d_out;
    tri_render_kernel<<<IH, 256, 0, stream>>>(verts, colors, depth, out);
}